// IWT3d_78348793413823
// MI455X (gfx1250) — compile-verified
//
#include <hip/hip_runtime.h>

// 3-D inverse wavelet recombine (8-band signed butterfly) for MI455X/gfx1250.
//
// Bandwidth-bound: 64 MiB in + 64 MiB out => ~5.8 us floor @ 23.3 TB/s.
// Each input element is loaded exactly once, each output stored exactly once,
// both fully coalesced. The per-voxel 8x8 +/-1 transform is executed on the
// matrix pipe via V_WMMA_F32_16X16X4_F32 (two K=4 accumulation steps), with
// the sign matrix synthesized in registers (A operand) and band data as B.

typedef __attribute__((ext_vector_type(2))) float v2f;
typedef __attribute__((ext_vector_type(8))) float v8f;

__device__ __forceinline__ float band_sign(int parity, int band) {
    // S = (-1)^(i*p + j*q + k*r) = (-1)^popcount(M & Bidx)
    return (__builtin_popcount(parity & band) & 1) ? -1.0f : 1.0f;
}

__global__ __launch_bounds__(256)
void iwt3d_wmma_kernel(const float* __restrict__ x, float* __restrict__ out) {
    const int tid  = (int)blockIdx.x * 256 + (int)threadIdx.x;
    const int lane = (int)threadIdx.x & 31;          // wave32
    const int wave = tid >> 5;

    // wave -> (b, c, d, h); each wave handles one full input W-row (32 voxels)
    const int h = wave & 31;
    const int d = (wave >> 5) & 31;
    const int c = (wave >> 10) & 31;
    const int b = (wave >> 15) & 1;

    const int bandStride = 32 * 32 * 32 * 32;        // elements between sub-bands (C*D*H*W)
    const int inBase = (((b * 256 + c) * 32 + d) * 32 + h) * 32;  // band 0, w = 0

    const int lh = lane & 15;                        // voxel column N / A row M
    const int hi = lane >> 4;                        // K high bit (lane-half striping)
    const int m  = lane & 7;                         // parity row (rows 8..15 duplicate)

    const int k0 = hi * 2;                           // K for VGPR0 of each A/B pair
    const int k1 = k0 + 1;                           // K for VGPR1

    // A operand: +/-1 sign matrix per the documented 16x4 f32 A layout.
    v2f A0, A1;
    A0.x = band_sign(m, k0);      A0.y = band_sign(m, k1);      // bands 0..3
    A1.x = band_sign(m, k0 + 4);  A1.y = band_sign(m, k1 + 4);  // bands 4..7

    // B operand: band data for 16 consecutive w. Mirrored K striping.
    v2f B00, B01, B10, B11;
    B00.x = x[inBase + (k0    ) * bandStride +  0 + lh];
    B00.y = x[inBase + (k1    ) * bandStride +  0 + lh];
    B01.x = x[inBase + (k0 + 4) * bandStride +  0 + lh];
    B01.y = x[inBase + (k1 + 4) * bandStride +  0 + lh];
    B10.x = x[inBase + (k0    ) * bandStride + 16 + lh];
    B10.y = x[inBase + (k1    ) * bandStride + 16 + lh];
    B11.x = x[inBase + (k0 + 4) * bandStride + 16 + lh];
    B11.y = x[inBase + (k1 + 4) * bandStride + 16 + lh];

    // D = A * B accumulated over the 8 bands (two K=4 WMMA steps per group).
    // EXEC is all-ones here (256-thread blocks, exact grid, no divergence yet).
    v8f acc0 = {};
    acc0 = __builtin_amdgcn_wmma_f32_16x16x4_f32(false, A0, false, B00,
                                                 (short)0, acc0, false, false);
    acc0 = __builtin_amdgcn_wmma_f32_16x16x4_f32(false, A1, false, B01,
                                                 (short)0, acc0, false, false);
    v8f acc1 = {};
    acc1 = __builtin_amdgcn_wmma_f32_16x16x4_f32(false, A0, false, B10,
                                                 (short)0, acc1, false, false);
    acc1 = __builtin_amdgcn_wmma_f32_16x16x4_f32(false, A1, false, B11,
                                                 (short)0, acc1, false, false);

    // D layout: lanes 0..15 hold voxel w0+lh, parities M=0..7 in VGPRs 0..7.
    // Parity M = 4p + 2q + r -> out[b][c][2d+p][2h+q][2w+r]; r=0,1 are the
    // two consecutive floats of one float2 store (128B contiguous/half-wave).
    if (lane < 16) {
        const int outRow = (((b * 32 + c) * 64 + 2 * d) * 64 + 2 * h) * 64;
        #pragma unroll
        for (int g = 0; g < 2; ++g) {
            const int w = g * 16 + lh;
            const v8f acc = g ? acc1 : acc0;
            const int outBase = outRow + 2 * w;
            #pragma unroll
            for (int p = 0; p < 2; ++p) {
                #pragma unroll
                for (int q = 0; q < 2; ++q) {
                    v2f v;
                    v.x = acc[4 * p + 2 * q + 0];   // r = 0
                    v.y = acc[4 * p + 2 * q + 1];   // r = 1
                    *(v2f*)(out + outBase + p * 4096 + q * 64) = v;
                }
            }
        }
    }
}

extern "C" void kernel_launch(void* const* d_in, const int* in_sizes, int n_in,
                              void* d_out, int out_size, void* d_ws, size_t ws_size,
                              hipStream_t stream) {
    (void)n_in; (void)out_size; (void)d_ws; (void)ws_size;
    const float* x = (const float*)d_in[0];
    float* out = (float*)d_out;

    // waves = B*C*D*H = total_elems / (8 bands * 32 w) ; 8 waves per block
    const int waves  = in_sizes[0] / 256;   // 65536 for (2,256,32,32,32)
    const int blocks = waves / 8;           // 8192 blocks of 256 threads
    hipLaunchKernelGGL(iwt3d_wmma_kernel, dim3(blocks), dim3(256), 0, stream,
                       x, out);
}